// MPGATLayer_85555748536493
// MI455X (gfx1250) — compile-verified
//
#include <hip/hip_runtime.h>
#include <math.h>

typedef float v2f __attribute__((ext_vector_type(2)));
typedef float v8f __attribute__((ext_vector_type(8)));

#define IN_F 128
#define OUT_F 64

// ---------------- init: zero accumulators (ws/out are poisoned by harness) --
__global__ void k_init(float* __restrict__ out, float* __restrict__ denom,
                       unsigned* __restrict__ emaxkey, int n) {
  int idx = blockIdx.x * blockDim.x + threadIdx.x;
  int total = n * OUT_F;
  if (idx < total) out[idx] = 0.0f;
  if (idx < n) {
    denom[idx] = 0.0f;
    emaxkey[idx] = 0u;  // minimal order-key (< key(-inf))
  }
}

// ---------------- GEMM: xv = x @ Wv + bv  via V_WMMA_F32_16X16X4_F32 -------
// block = 256 threads = 8 waves; block tile = 32 rows x 64 cols,
// one 16x16 C tile per wave, K swept in steps of 4.
__global__ __launch_bounds__(256) void k_gemm_xv(
    const float* __restrict__ x, const float* __restrict__ Wv,
    const float* __restrict__ bv, float* __restrict__ xv) {
  __shared__ float As[32 * 132];   // 32 rows x 128 K, padded to 132
  __shared__ float Bs[128 * 65];   // 128 K x 64 N, padded to 65

  const int t = threadIdx.x;
  const int row0blk = blockIdx.x * 32;

  // stage Wv -> LDS (8192 floats)
  for (int i = t; i < IN_F * OUT_F; i += 256) {
    int r = i >> 6, c = i & 63;
    Bs[r * 65 + c] = Wv[i];
  }
  // stage x tile -> LDS (32 rows x 128 = 1024 float4, coalesced b128 loads)
  for (int i = t; i < 32 * 32; i += 256) {
    int r = i >> 5, c4 = i & 31;
    const float4 v =
        ((const float4*)(x + (size_t)(row0blk + r) * IN_F))[c4];
    *(float4*)(&As[r * 132 + c4 * 4]) = v;
  }
  __syncthreads();

  const int wave = t >> 5;
  const int lane = t & 31;
  const int mn = lane & 15;        // A row (m) / B,C col (n) within tile
  const int half = lane >> 4;      // selects K pair (A/B) and M half (C)
  const int waveM = wave >> 2;     // 0..1  -> M tile
  const int n0 = (wave & 3) * 16;  // 0,16,32,48 -> N tile

  v8f c = {0.f, 0.f, 0.f, 0.f, 0.f, 0.f, 0.f, 0.f};
  const float* Arow = &As[(waveM * 16 + mn) * 132];

#pragma unroll
  for (int k0 = 0; k0 < IN_F; k0 += 4) {
    const int kk = k0 + 2 * half;
    v2f a, b;
    // A 16x4 layout: lanes 0-15 hold K={k0,k0+1}, lanes 16-31 K={k0+2,k0+3}
    a.x = Arow[kk];
    a.y = Arow[kk + 1];
    // B 4x16 layout: lane = n, VGPR0/1 hold the two K values of this half
    b.x = Bs[kk * 65 + n0 + mn];
    b.y = Bs[(kk + 1) * 65 + n0 + mn];
    c = __builtin_amdgcn_wmma_f32_16x16x4_f32(false, a, false, b, (short)0, c,
                                              false, false);
  }

  // C 16x16 layout: VGPR i -> M = i + 8*half, N = lane&15
  const float bias = bv[n0 + mn];
  const int row0 = row0blk + waveM * 16 + half * 8;
#pragma unroll
  for (int i = 0; i < 8; ++i) {
    xv[(size_t)(row0 + i) * OUT_F + n0 + mn] = c[i] + bias;
  }
}

// ---------------- el/er: per-node dot products, wave32 shuffle reduction ---
__global__ __launch_bounds__(256) void k_scores(
    const float* __restrict__ xv, const float* __restrict__ wq,
    const float* __restrict__ bq, const float* __restrict__ wk,
    const float* __restrict__ bk, float* __restrict__ el,
    float* __restrict__ er, int n) {
  int node = blockIdx.x * 8 + (threadIdx.x >> 5);
  int lane = threadIdx.x & 31;
  if (node >= n) return;
  float2 v = ((const float2*)(xv + (size_t)node * OUT_F))[lane];
  float2 q = ((const float2*)wq)[lane];
  float2 k2 = ((const float2*)wk)[lane];
  float dq = v.x * q.x + v.y * q.y;
  float dk = v.x * k2.x + v.y * k2.y;
#pragma unroll
  for (int off = 16; off >= 1; off >>= 1) {
    dq += __shfl_xor(dq, off, 32);
    dk += __shfl_xor(dk, off, 32);
  }
  if (lane == 0) {
    el[node] = dq + bq[0];
    er[node] = dk + bk[0];
  }
}

// order-preserving float<->uint key for atomic segment-max
__device__ __forceinline__ unsigned f2key(float f) {
  unsigned u = __float_as_uint(f);
  return (u & 0x80000000u) ? ~u : (u | 0x80000000u);
}
__device__ __forceinline__ float key2f(unsigned k) {
  return (k & 0x80000000u) ? __uint_as_float(k ^ 0x80000000u)
                           : __uint_as_float(~k);
}

// ---------------- edge pass 1: leaky_relu score + segment max --------------
__global__ void k_edge_max(const long long* __restrict__ src,
                           const long long* __restrict__ dst,
                           const float* __restrict__ el,
                           const float* __restrict__ er,
                           float* __restrict__ ebuf,
                           unsigned* __restrict__ emaxkey, int ne) {
  int e = blockIdx.x * blockDim.x + threadIdx.x;
  if (e >= ne) return;
  int s = (int)src[e], d = (int)dst[e];
  float v = el[s] + er[d];
  v = v > 0.0f ? v : 0.2f * v;
  ebuf[e] = v;
  atomicMax(&emaxkey[d], f2key(v));
}

// ---------------- edge pass 2: exp(e - max) + segment sum ------------------
__global__ void k_edge_exp(const long long* __restrict__ dst,
                           const unsigned* __restrict__ emaxkey,
                           float* __restrict__ ebuf,
                           float* __restrict__ denom, int ne) {
  int e = blockIdx.x * blockDim.x + threadIdx.x;
  if (e >= ne) return;
  int d = (int)dst[e];
  float ex = __expf(ebuf[e] - key2f(emaxkey[d]));
  ebuf[e] = ex;
  atomicAdd(&denom[d], ex);
}

// ---------------- edge pass 3: normalize + weighted scatter ----------------
// 16 lanes per edge: float4 gather of xv[src], 4 f32 atomics into out[dst]
__global__ __launch_bounds__(256) void k_scatter(
    const long long* __restrict__ src, const long long* __restrict__ dst,
    const float* __restrict__ xv, const float* __restrict__ ebuf,
    const float* __restrict__ denom, float* __restrict__ out, int ne) {
  int e = (int)((blockIdx.x * 256 + threadIdx.x) >> 4);
  int l = threadIdx.x & 15;
  if (e >= ne) return;
  int s = (int)src[e], d = (int)dst[e];
  float attn = ebuf[e] / denom[d];
  float4 v = ((const float4*)(xv + (size_t)s * OUT_F))[l];
  float* o = out + (size_t)d * OUT_F + l * 4;
  atomicAdd(o + 0, v.x * attn);
  atomicAdd(o + 1, v.y * attn);
  atomicAdd(o + 2, v.z * attn);
  atomicAdd(o + 3, v.w * attn);
}

extern "C" void kernel_launch(void* const* d_in, const int* in_sizes, int n_in,
                              void* d_out, int out_size, void* d_ws,
                              size_t ws_size, hipStream_t stream) {
  const float* x = (const float*)d_in[0];
  const float* Wv = (const float*)d_in[1];
  const float* bv = (const float*)d_in[2];
  const float* wq = (const float*)d_in[3];
  const float* bq = (const float*)d_in[4];
  const float* wk = (const float*)d_in[5];
  const float* bk = (const float*)d_in[6];
  const long long* src = (const long long*)d_in[7];
  const long long* dst = (const long long*)d_in[8];
  float* out = (float*)d_out;

  const int n = in_sizes[0] / IN_F;  // 100000
  const int ne = in_sizes[7];        // 1600000

  // workspace carve-up (~33.6 MB total)
  char* ws = (char*)d_ws;
  float* xv = (float*)ws;            ws += (size_t)n * OUT_F * sizeof(float);
  float* el = (float*)ws;            ws += (size_t)n * sizeof(float);
  float* er = (float*)ws;            ws += (size_t)n * sizeof(float);
  unsigned* emaxkey = (unsigned*)ws; ws += (size_t)n * sizeof(unsigned);
  float* denom = (float*)ws;         ws += (size_t)n * sizeof(float);
  float* ebuf = (float*)ws;

  k_init<<<(n * OUT_F + 255) / 256, 256, 0, stream>>>(out, denom, emaxkey, n);
  k_gemm_xv<<<(n + 31) / 32, 256, 0, stream>>>(x, Wv, bv, xv);
  k_scores<<<(n + 7) / 8, 256, 0, stream>>>(xv, wq, bq, wk, bk, el, er, n);
  k_edge_max<<<(ne + 255) / 256, 256, 0, stream>>>(src, dst, el, er, ebuf,
                                                   emaxkey, ne);
  k_edge_exp<<<(ne + 255) / 256, 256, 0, stream>>>(dst, emaxkey, ebuf, denom,
                                                   ne);
  k_scatter<<<(ne * 16 + 255) / 256, 256, 0, stream>>>(src, dst, xv, ebuf,
                                                       denom, out, ne);
}